// LGCN_Encoder_46952582479983
// MI455X (gfx1250) — compile-verified
//
#include <hip/hip_runtime.h>

#define EMB 64

typedef __attribute__((ext_vector_type(2))) float v2f;
typedef __attribute__((ext_vector_type(8))) float v8f;

#if defined(__has_builtin)
#if __has_builtin(__builtin_amdgcn_wmma_f32_16x16x4_f32)
#define HAVE_WMMA_F32K4 1
#endif
#endif

// ---------------------------------------------------------------------------
// out = 0.25*ego ; x = ego   (ego = concat(user_emb, item_emb)), float4-wide
// ---------------------------------------------------------------------------
__global__ void __launch_bounds__(256)
lgcn_init(const float4* __restrict__ u, const float4* __restrict__ it,
          float4* __restrict__ x, float4* __restrict__ out, int nu4, int n4) {
  int i = blockIdx.x * blockDim.x + threadIdx.x;
  if (i < n4) {
    float4 v = (i < nu4) ? u[i] : it[i - nu4];
    x[i] = v;
    float4 o;
    o.x = 0.25f * v.x; o.y = 0.25f * v.y; o.z = 0.25f * v.z; o.w = 0.25f * v.w;
    out[i] = o;
  }
}

__global__ void __launch_bounds__(256)
lgcn_zero(float4* __restrict__ p, int n4) {
  int i = blockIdx.x * blockDim.x + threadIdx.x;
  if (i < n4) { float4 z; z.x = z.y = z.z = z.w = 0.0f; p[i] = z; }
}

// ---------------------------------------------------------------------------
// xnext = (1+c)*e - c*m ; out += 0.25*xnext
// ---------------------------------------------------------------------------
__global__ void __launch_bounds__(256)
lgcn_combine(const float4* __restrict__ e, const float4* __restrict__ m,
             float4* __restrict__ xn, float4* __restrict__ out,
             float coef, int n4) {
  int i = blockIdx.x * blockDim.x + threadIdx.x;
  if (i < n4) {
    float a = 1.0f + coef, b = coef;
    float4 ev = e[i], mv = m[i], v, o = out[i];
    v.x = a * ev.x - b * mv.x;  v.y = a * ev.y - b * mv.y;
    v.z = a * ev.z - b * mv.z;  v.w = a * ev.w - b * mv.w;
    xn[i] = v;
    o.x += 0.25f * v.x; o.y += 0.25f * v.y;
    o.z += 0.25f * v.z; o.w += 0.25f * v.w;
    out[i] = o;
  }
}

// ---------------------------------------------------------------------------
// COO SpMM: out[rows[i]] += vals[i] * x[cols[i]], 16 nnz per wave32 batch.
// D(16x16) = diag(v) * G(16x16 gathered tile) on the matrix pipe via chained
// V_WMMA_F32_16X16X4_F32 (K=16 as 4 chained K=4), scattered with f32 L2
// atomics using the C/D register layout:
//   lane l holds D[t + 8*(l>>4)][l&15] in acc[t].
// Layouts (16x4 A / 4x16 B, 32-bit, wave32):
//   A: a.x = A[m][2*hf], a.y = A[m][2*hf+1]     (m = lane&15, hf = lane>>4)
//   B: b.x = B[2*hf][n], b.y = B[2*hf+1][n]     (n = lane&15)
// Row shuffles + scatter offsets are hoisted out of the column-tile loop:
// the 8 destination rows are identical across all four 16-col tiles.
// ---------------------------------------------------------------------------
__global__ void __launch_bounds__(256)
lgcn_spmm_wmma(const int* __restrict__ rows, const int* __restrict__ cols,
               const float* __restrict__ vals, const float* __restrict__ x,
               float* __restrict__ out, int nbatch) {
  const int lane = threadIdx.x & 31;
  const int wave = blockIdx.x * (blockDim.x >> 5) + (threadIdx.x >> 5);
  const int nwaves = gridDim.x * (blockDim.x >> 5);
  const int m  = lane & 15;
  const int hf = lane >> 4;
  const int n  = m;

  for (int bb = wave; bb < nbatch; bb += nwaves) {
    const int base = bb << 4;

    // Prefetch the next COO batch this wave will touch (predictable stream).
    if (bb + nwaves < nbatch) {
      const int nb = (bb + nwaves) << 4;
      __builtin_prefetch(cols + nb, 0, 1);
      __builtin_prefetch(rows + nb, 0, 1);
      __builtin_prefetch(vals + nb, 0, 1);
    }

    const float vm = vals[base + m];   // value of entry m (dup across halves)
    const int   rm = rows[base + m];   // dest row of entry m

    // Scatter offsets (element index into out), same for all 4 column tiles.
    int soff[8];
#pragma unroll
    for (int t = 0; t < 8; ++t) {
      const int r = __shfl(rm, t + 8 * hf);   // rows[base + t + 8*hf]
      soff[t] = r * EMB + n;
    }

#if defined(HAVE_WMMA_F32K4)
    // Column indices feeding the B tiles: chunk kk uses entries 4kk+{2hf,2hf+1}
    int c0[4], c1[4];
#pragma unroll
    for (int kk = 0; kk < 4; ++kk) {
      c0[kk] = cols[base + 4 * kk + 2 * hf] * EMB + n;
      c1[kk] = cols[base + 4 * kk + 2 * hf + 1] * EMB + n;
    }
    // A chunks: diag(v) sliced into four 16x4 blocks
    v2f a[4];
#pragma unroll
    for (int kk = 0; kk < 4; ++kk) {
      const int k0 = 4 * kk + 2 * hf;
      a[kk].x = (m == k0)     ? vm : 0.0f;
      a[kk].y = (m == k0 + 1) ? vm : 0.0f;
    }

#pragma unroll
    for (int jt = 0; jt < 4; ++jt) {       // four 16-column tiles of EMB=64
      const int jb = jt * 16;
      v8f acc = {};
#pragma unroll
      for (int kk = 0; kk < 4; ++kk) {
        v2f b;
        b.x = x[c0[kk] + jb];
        b.y = x[c1[kk] + jb];
        acc = __builtin_amdgcn_wmma_f32_16x16x4_f32(
            false, a[kk], false, b, (short)0, acc, false, false);
      }
#pragma unroll
      for (int t = 0; t < 8; ++t)
        atomicAdd(&out[soff[t] + jb], acc[t]);
    }
#else
    // VALU fallback: D[mrow][n] = v_mrow * x[c_mrow][jb+n]
    const int cm = cols[base + m];
    float vv[8]; int cc[8];
#pragma unroll
    for (int t = 0; t < 8; ++t) {
      vv[t] = __shfl(vm, t + 8 * hf);
      cc[t] = __shfl(cm, t + 8 * hf) * EMB + n;
    }
#pragma unroll
    for (int jt = 0; jt < 4; ++jt) {
      const int jb = jt * 16;
#pragma unroll
      for (int t = 0; t < 8; ++t)
        atomicAdd(&out[soff[t] + jb], vv[t] * x[cc[t] + jb]);
    }
#endif
  }
}

// Tail for nnz not a multiple of 16 (not hit for NNZ=1M; kept for generality).
__global__ void __launch_bounds__(256)
lgcn_spmm_tail(const int* __restrict__ rows, const int* __restrict__ cols,
               const float* __restrict__ vals, const float* __restrict__ x,
               float* __restrict__ out, int start, int nnz) {
  int t = blockIdx.x * blockDim.x + threadIdx.x;
  int total = (nnz - start) * EMB;
  if (t < total) {
    int e = start + t / EMB;
    int j = t % EMB;
    atomicAdd(&out[rows[e] * EMB + j], vals[e] * x[cols[e] * EMB + j]);
  }
}

// ---------------------------------------------------------------------------
extern "C" void kernel_launch(void* const* d_in, const int* in_sizes, int n_in,
                              void* d_out, int out_size, void* d_ws, size_t ws_size,
                              hipStream_t stream) {
  const float* user_emb = (const float*)d_in[0];
  const float* item_emb = (const float*)d_in[1];
  const int*   Rm[3] = {(const int*)d_in[2], (const int*)d_in[5], (const int*)d_in[8]};
  const int*   Cm[3] = {(const int*)d_in[3], (const int*)d_in[6], (const int*)d_in[9]};
  const float* Vm[3] = {(const float*)d_in[4], (const float*)d_in[7], (const float*)d_in[10]};
  const int nnz[3] = {in_sizes[2], in_sizes[5], in_sizes[8]};

  const int ntot = in_sizes[0] + in_sizes[1];   // N * EMB elements
  const int nu4  = in_sizes[0] / 4;
  const int n4   = ntot / 4;

  float* out = (float*)d_out;
  float* x  = (float*)d_ws;          // rotating 38.4 MB buffers in workspace
  float* t1 = x  + ntot;
  float* t2 = t1 + ntot;

  const dim3 blk(256);
  const dim3 grdE((n4 + 255) / 256);
  const dim3 grdS(768);              // 6144 wave32's over 62500 nnz batches

  auto spmm = [&](const int* r, const int* c, const float* v,
                  const float* xin, float* yout, int m) {
    lgcn_zero<<<grdE, blk, 0, stream>>>((float4*)yout, n4);
    const int nbatch = m >> 4;
    if (nbatch > 0)
      lgcn_spmm_wmma<<<grdS, blk, 0, stream>>>(r, c, v, xin, yout, nbatch);
    const int rem = m & 15;
    if (rem > 0)
      lgcn_spmm_tail<<<dim3((rem * EMB + 255) / 256), blk, 0, stream>>>(
          r, c, v, xin, yout, m - rem, m);
  };

  lgcn_init<<<grdE, blk, 0, stream>>>((const float4*)user_emb,
                                      (const float4*)item_emb,
                                      (float4*)x, (float4*)out, nu4, n4);

  const float coefs[3] = {0.1f, 0.1f, 0.0f};
  for (int h = 0; h < 3; ++h) {
    spmm(Rm[0], Cm[0], Vm[0], x,  t1, nnz[0]);   // e  = A  * x
    spmm(Rm[1], Cm[1], Vm[1], t1, t2, nnz[1]);   // m1 = AJ * e
    spmm(Rm[2], Cm[2], Vm[2], t2, x,  nnz[2]);   // m2 = AI * m1 (overwrites x)
    // xnext = (1+c)*e - c*m2 -> t2 ; out += 0.25*xnext
    lgcn_combine<<<grdE, blk, 0, stream>>>((const float4*)t1, (const float4*)x,
                                           (float4*)t2, (float4*)out,
                                           coefs[h], n4);
    float* nx = t2; t2 = t1; t1 = x; x = nx;     // rotate buffers
  }
}